// DigitCapsules_88991722373499
// MI455X (gfx1250) — compile-verified
//
#include <hip/hip_runtime.h>
#include <hip/hip_bf16.h>

// DigitCaps dynamic routing for MI455X (gfx1250, wave32).
// - fp32 WMMA 16x16x4 for u_hat = x @ W per n (full fp32 precision; the
//   GEMM is store-BW bound at ~4 FLOP/byte so WMMA trivially keeps up).
// - u_hat (189 MB) materialized once; fits the 192 MB L2, re-read 3x.
// - Routing: thread<->(o,k), shfl k-reductions, LDS softmax over o (10 exps
//   per (b,n)), single barrier per n (parity-buffered eL), software-pipelined
//   u loads + global_prefetch_b8.
// - Deterministic: no float atomics; per-chunk s partials reduced in fixed
//   order by the finalize kernel.

#define NI 1152
#define BB 256
#define OO 10
#define IL 8
#define OL 16
#define OK (OO * OL)      // 160 columns per (b,n)
#define NCHUNK 128        // n per routing block; 1152/128 = 9 chunks
#define CHUNKS (NI / NCHUNK)

typedef float v2f __attribute__((ext_vector_type(2)));
typedef float v8f __attribute__((ext_vector_type(8)));

// ---------------------------------------------------------------------------
// Kernel A: u_hat[b,n,o,k] = sum_i W[n,o,i,k] * x[b,n,i]
// One block per n: C (256x160) = X_n (256x8) @ W_n (8x160).
// 8 waves; wave w owns M-tiles {2w,2w+1}; 10 o-tiles x 2 WMMA k-steps.
// Stores use a single per-thread base pointer + compile-time element offsets
// (r*NI*OK + o*OL; max byte offset 5.16 MB < 2^23) so the backend folds them
// into global_store_b32 immediate offsets instead of per-store 64-bit mads.
// ---------------------------------------------------------------------------
__global__ __launch_bounds__(256) void uhat_wmma_kernel(
    const float* __restrict__ x, const float* __restrict__ w,
    float* __restrict__ u) {
  const int n = blockIdx.x;
  __shared__ float Xs[BB * IL];       // 8 KB: X_n, row b, 8 floats
  __shared__ float Ws[OO * IL * OL];  // 5 KB: W_n, [o][i][k]
  const int t = threadIdx.x;

  // Load X_n: thread t loads row b=t (8 contiguous floats, rows 36KB apart).
  {
    const float* xp = x + (size_t)t * (NI * IL) + (size_t)n * IL;
#pragma unroll
    for (int i = 0; i < IL; ++i) Xs[t * IL + i] = xp[i];
  }
  // Load W_n: 1280 contiguous floats, fully coalesced.
  for (int j = t; j < OO * IL * OL; j += 256)
    Ws[j] = w[(size_t)n * (OO * IL * OL) + j];
  __syncthreads();

  const int wv  = t >> 5;       // wave id 0..7
  const int ln  = t & 31;
  const int lhi = ln >> 4;      // 0: lanes 0-15, 1: lanes 16-31
  const int llo = ln & 15;

  // A fragments (16x4 f32): VGPR j holds K = 2*lhi + j (+4 per k-step).
  v2f afr[2][2];
#pragma unroll
  for (int mt = 0; mt < 2; ++mt) {
    const int m = (2 * wv + mt) * 16 + llo;
#pragma unroll
    for (int ks = 0; ks < 2; ++ks) {
      const int i0 = ks * 4 + lhi * 2;
      afr[mt][ks].x = Xs[m * IL + i0];
      afr[mt][ks].y = Xs[m * IL + i0 + 1];
    }
  }

  // Base store pointers: D VGPR r -> row (mtile*16 + lhi*8 + r), col llo.
  float* up0 = u + ((size_t)((2 * wv + 0) * 16 + lhi * 8) * NI + n) * OK + llo;
  float* up1 = u + ((size_t)((2 * wv + 1) * 16 + lhi * 8) * NI + n) * OK + llo;

#pragma unroll
  for (int o = 0; o < OO; ++o) {
    // B fragments (4x16 f32): rows 2*lhi(+1) per k-step, col llo.
    v2f bfr[2];
#pragma unroll
    for (int ks = 0; ks < 2; ++ks) {
      const int i0 = ks * 4 + lhi * 2;
      bfr[ks].x = Ws[o * (IL * OL) + i0 * OL + llo];
      bfr[ks].y = Ws[o * (IL * OL) + (i0 + 1) * OL + llo];
    }
    v8f acc0 = {}, acc1 = {};
    acc0 = __builtin_amdgcn_wmma_f32_16x16x4_f32(false, afr[0][0], false,
                                                 bfr[0], (short)0, acc0,
                                                 false, false);
    acc0 = __builtin_amdgcn_wmma_f32_16x16x4_f32(false, afr[0][1], false,
                                                 bfr[1], (short)0, acc0,
                                                 false, false);
    acc1 = __builtin_amdgcn_wmma_f32_16x16x4_f32(false, afr[1][0], false,
                                                 bfr[0], (short)0, acc1,
                                                 false, false);
    acc1 = __builtin_amdgcn_wmma_f32_16x16x4_f32(false, afr[1][1], false,
                                                 bfr[1], (short)0, acc1,
                                                 false, false);
#pragma unroll
    for (int r = 0; r < 8; ++r) {
      up0[(size_t)r * (NI * OK) + o * OL] = acc0[r];
      up1[(size_t)r * (NI * OK) + o * OL] = acc1[r];
    }
  }
}

// ---------------------------------------------------------------------------
// Routing pass. Grid (CHUNKS, B), 160 threads: t -> (o = t/16, k = t%16).
// iter 0: c = 1/10 (b_ij == 0). iter >= 1: agreement with v_prev, b_ij
// update, softmax over o. One barrier per n (eL parity double-buffer).
// ---------------------------------------------------------------------------
__global__ __launch_bounds__(160) void route_kernel(
    const float* __restrict__ u, const float* __restrict__ vprev,
    float* __restrict__ bij, float* __restrict__ spart, int iter) {
  const int chunk = blockIdx.x;
  const int b = blockIdx.y;
  const int t = threadIdx.x;
  const int o = t >> 4;

  __shared__ float bL[NCHUNK * OO];  // b_ij slice, 5 KB
  __shared__ float eL[2][OO];        // exp(b) per n, parity-buffered
  __shared__ float vL[OK];           // previous v for this b

  if (iter > 0) {
    vL[t] = vprev[b * OK + t];
    float* bg = bij + (size_t)b * (NI * OO) + (size_t)chunk * (NCHUNK * OO);
    for (int j = t; j < NCHUNK * OO; j += 160) bL[j] = bg[j];
  }
  __syncthreads();

  float sacc = 0.f;
  const float* ub = u + ((size_t)b * NI + (size_t)chunk * NCHUNK) * OK;

  float uu = ub[t];  // software-pipelined u load
  for (int nl = 0; nl < NCHUNK; ++nl) {
    float un = 0.f;
    if (nl + 1 < NCHUNK) un = ub[(nl + 1) * OK + t];
    {
      const int pf = (nl + 8 < NCHUNK) ? (nl + 8) : (NCHUNK - 1);
      __builtin_prefetch(ub + pf * OK + t, 0, 1);  // global_prefetch_b8
    }
    float c;
    if (iter > 0) {
      // agreement a[o] = sum_k u*v : 16-lane butterfly within wave halves
      float p = uu * vL[t];
      p += __shfl_xor(p, 1, 32);
      p += __shfl_xor(p, 2, 32);
      p += __shfl_xor(p, 4, 32);
      p += __shfl_xor(p, 8, 32);
      if ((t & 15) == 0) {
        const float bnew = bL[nl * OO + o] + p;
        bL[nl * OO + o] = bnew;          // only this lane rereads it
        eL[nl & 1][o] = __expf(bnew);    // b_ij stays O(1); no max-shift
      }
      __syncthreads();  // parity buffer: no barrier needed after the reads
      float sum = 0.f;
#pragma unroll
      for (int j = 0; j < OO; ++j) sum += eL[nl & 1][j];
      c = eL[nl & 1][o] / sum;
    } else {
      c = 0.1f;  // softmax of zeros over O=10
    }
    sacc += c * uu;
    uu = un;
  }

  if (iter > 0) {
    float* bg = bij + (size_t)b * (NI * OO) + (size_t)chunk * (NCHUNK * OO);
    for (int j = t; j < NCHUNK * OO; j += 160) bg[j] = bL[j];
  }
  // deterministic partial: fixed (chunk,b) slot, no atomics
  spart[((size_t)chunk * BB + b) * OK + t] = sacc;
}

// ---------------------------------------------------------------------------
// Finalize: s = bias + sum_chunks spart (fixed order); v = squash(s).
// Grid B, 160 threads. Norm over k via the same 16-lane butterfly.
// ---------------------------------------------------------------------------
__global__ __launch_bounds__(160) void finalize_kernel(
    const float* __restrict__ spart, const float* __restrict__ bias,
    float* __restrict__ vout) {
  const int b = blockIdx.x;
  const int t = threadIdx.x;
  float s = bias[t];
#pragma unroll
  for (int ch = 0; ch < CHUNKS; ++ch)
    s += spart[((size_t)ch * BB + b) * OK + t];
  float d = s * s;
  d += __shfl_xor(d, 1, 32);
  d += __shfl_xor(d, 2, 32);
  d += __shfl_xor(d, 4, 32);
  d += __shfl_xor(d, 8, 32);
  const float scale = d / ((1.f + d) * sqrtf(d + 1e-7f));
  vout[b * OK + t] = s * scale;
}

__global__ void zero_kernel(float* __restrict__ p, int nfl) {
  const int i = blockIdx.x * blockDim.x + threadIdx.x;
  if (i < nfl) p[i] = 0.f;
}

// ---------------------------------------------------------------------------
extern "C" void kernel_launch(void* const* d_in, const int* in_sizes, int n_in,
                              void* d_out, int out_size, void* d_ws,
                              size_t ws_size, hipStream_t stream) {
  const float* x    = (const float*)d_in[0];  // (256,1152,8,1)
  const float* w    = (const float*)d_in[1];  // (1,1152,10,8,16)
  const float* bias = (const float*)d_in[2];  // (1,1,10,16,1) -> 160 floats

  // Workspace layout (floats): u_hat | b_ij | spart | vbuf  (~202 MB total)
  float* u    = (float*)d_ws;                    // B*N*160
  float* bij  = u + (size_t)BB * NI * OK;        // B*N*10
  float* sp   = bij + (size_t)BB * NI * OO;      // CHUNKS*B*160
  float* vbuf = sp + (size_t)CHUNKS * BB * OK;   // B*160
  float* vout = (float*)d_out;                   // B*10*16

  // b_ij = 0
  {
    const int nfl = BB * NI * OO;
    zero_kernel<<<(nfl + 255) / 256, 256, 0, stream>>>(bij, nfl);
  }
  // u_hat via WMMA
  uhat_wmma_kernel<<<NI, 256, 0, stream>>>(x, w, u);

  // 3 routing iterations; last finalize writes d_out
  dim3 rgrid(CHUNKS, BB);
  route_kernel<<<rgrid, 160, 0, stream>>>(u, vbuf, bij, sp, 0);
  finalize_kernel<<<BB, 160, 0, stream>>>(sp, bias, vbuf);
  route_kernel<<<rgrid, 160, 0, stream>>>(u, vbuf, bij, sp, 1);
  finalize_kernel<<<BB, 160, 0, stream>>>(sp, bias, vbuf);
  route_kernel<<<rgrid, 160, 0, stream>>>(u, vbuf, bij, sp, 2);
  finalize_kernel<<<BB, 160, 0, stream>>>(sp, bias, vout);
}